// AutoEncoder_60550448939424
// MI455X (gfx1250) — compile-verified
//
#include <hip/hip_runtime.h>
#include <hip/hip_bf16.h>
#include <math.h>

// Problem constants (from reference)
#define B_  512
#define N_  10020
#define L_  20
#define S_  10
#define H_  1001   // (N-L)/S + 1
#define INV_SQRT6 0.40824829046386301636f

typedef __attribute__((ext_vector_type(2))) float v2f;
typedef __attribute__((ext_vector_type(8))) float v8f;

// ---------------------------------------------------------------------------
// Kernel 1: contribution (N,H) and nw (N,H). One wave32 per row n.
// ---------------------------------------------------------------------------
__global__ __launch_bounds__(256) void nw_kernel(const float* __restrict__ Wenc,
                                                 float* __restrict__ nw_out,
                                                 float* __restrict__ contrib_out) {
  const int row  = blockIdx.x * (blockDim.x >> 5) + (threadIdx.x >> 5);
  if (row >= N_) return;
  const int lane = threadIdx.x & 31;
  // started = ceil((row+1)/S); last L rows clamp to started[N-L]
  const int started = (row >= N_ - L_) ? ((N_ - L_ + 1 + S_ - 1) / S_)
                                       : ((row + 1 + S_ - 1) / S_);
  // ended = max(ceil((row+1-L)/S), 0)
  const int ended = (row + 1 > L_) ? ((row + 1 - L_ + S_ - 1) / S_) : 0;

  float sum = 0.f;
  for (int h = lane; h < H_; h += 32) {
    const float c = (h >= ended && h < started) ? 1.f : 0.f;
    contrib_out[(size_t)row * H_ + h] = c;
    const int shift = row - S_ * h;
    float v = 0.f;
    if (shift >= 0 && shift < L_) v = c * Wenc[((size_t)shift * H_ + h) * 3];
    sum += v;
  }
  #pragma unroll
  for (int off = 16; off >= 1; off >>= 1) sum += __shfl_xor(sum, off, 32);
  for (int h = lane; h < H_; h += 32) {
    const float c = (h >= ended && h < started) ? 1.f : 0.f;
    const int shift = row - S_ * h;
    float v = 0.f;
    if (shift >= 0 && shift < L_) v = c * Wenc[((size_t)shift * H_ + h) * 3];
    nw_out[(size_t)row * H_ + h] = v / sum;
  }
}

// ---------------------------------------------------------------------------
// Kernel 2: st[b,i,h,k] = x[b, S*h + i, k].  Coalesced on the output.
// ---------------------------------------------------------------------------
__global__ __launch_bounds__(256) void st_kernel(const float* __restrict__ x,
                                                 float* __restrict__ st_out) {
  const long long i = (long long)blockIdx.x * blockDim.x + threadIdx.x;
  const long long total = (long long)B_ * L_ * H_ * 3;
  if (i >= total) return;
  const int k = (int)(i % 3);
  long long t = i / 3;
  const int h = (int)(t % H_); t /= H_;
  const int li = (int)(t % L_);
  const int b = (int)(t / L_);
  st_out[i] = x[((size_t)b * N_ + (size_t)(S_ * h + li)) * 3 + k];
}

// ---------------------------------------------------------------------------
// Kernel 3: encode. One wave per (b-tile of 16, h).
//   emb[b, n=l*3+k] = sum_{kk<60} A[b,kk] * Bmat[kk][n]
//     A[m,kk]     = x[(b0+m)*N*3 + 30*h + kk]          (contiguous in kk)
//     Bmat[kk][n] = (kk%3==n%3) ? W_enc[(kk/3)*H+h)*3 + n/3] : 0   (n<9)
// B is staged in LDS (one 60x16 panel per wave) so the K-loop is pure
//   global_load_b64 (A) + ds_load_b64 (B) + V_WMMA_F32_16X16X4_F32.
// Then P/p1/p2/p12 per b, and rot_axes = [p1,p2,p12] · W_dec.
// ---------------------------------------------------------------------------
__global__ __launch_bounds__(256) void encode_kernel(const float* __restrict__ x,
                                                     const float* __restrict__ Wenc,
                                                     const float* __restrict__ Wdec,
                                                     float* __restrict__ P_out,
                                                     float* __restrict__ p1_out,
                                                     float* __restrict__ p2_out,
                                                     float* __restrict__ rot_out) {
  __shared__ float Blds[8][16][68];   // [wave][n][kk], 68-row pad (bank spread)
  __shared__ float embS[8][16][12];   // [wave][M=b-local][n<9]
  __shared__ float vecS[8][16][12];   // [wave][b-local][p1|p2|p12]

  const int wv   = threadIdx.x >> 5;
  const int lane = threadIdx.x & 31;
  const int task = blockIdx.x * 8 + wv;         // grid sized exactly 32*H/8
  const int h    = task % H_;
  const int b0   = (task / H_) * 16;

  const int m = lane & 15;                      // M row (A) / N col (B)
  const int o = (lane < 16) ? 0 : 2;            // K-pair select (ISA A layout)
  const float* xrow = x + (size_t)(b0 + m) * (N_ * 3) + (size_t)(3 * S_) * h;

  // ---- stage B panel in LDS: zero, then scatter the block-diagonal entries
  {
    float* bflat = &Blds[wv][0][0];
    for (int t = lane; t < 16 * 68; t += 32) bflat[t] = 0.f;
  }
  if (lane < 9) {
    const int kn = lane % 3;                    // component (k of n=l*3+k)
    const int ln = lane / 3;                    // output vector index l
    #pragma unroll
    for (int i = 0; i < L_; ++i)
      Blds[wv][lane][3 * i + kn] = Wenc[((size_t)i * H_ + h) * 3 + ln];
  }
  __syncthreads();

  union { v8f v; float f[8]; } acc;
  #pragma unroll
  for (int r = 0; r < 8; ++r) acc.f[r] = 0.f;

  const float* bcol = &Blds[wv][m][0];
  #pragma unroll
  for (int c = 0; c < 15; ++c) {
    const int kk = 4 * c + o;
    v2f a;  a.x  = xrow[kk];  a.y  = xrow[kk + 1];   // A frag (b64 load)
    v2f bb; bb.x = bcol[kk];  bb.y = bcol[kk + 1];   // B frag (ds b64 load)
    acc.v = __builtin_amdgcn_wmma_f32_16x16x4_f32(
        /*neg_a=*/false, a, /*neg_b=*/false, bb,
        /*c_mod=*/(short)0, acc.v, /*reuse_a=*/false, /*reuse_b=*/false);
  }

  // C/D layout: lane t, VGPR r  <->  (M = r + (t>=16 ? 8 : 0), N = t%16)
  if (m < 9) {
    const int mb = (lane < 16) ? 0 : 8;
    #pragma unroll
    for (int r = 0; r < 8; ++r) embS[wv][mb + r][m] = acc.f[r];
  }
  __syncthreads();

  if (lane < 16) {
    float P[3], d1[3], d2[3];
    #pragma unroll
    for (int k = 0; k < 3; ++k) {
      P[k]  = embS[wv][lane][k];
      d1[k] = embS[wv][lane][3 + k] - P[k];
      d2[k] = embS[wv][lane][6 + k] - P[k];
    }
    float n1 = sqrtf(d1[0]*d1[0] + d1[1]*d1[1] + d1[2]*d1[2]);
    float n2 = sqrtf(d2[0]*d2[0] + d2[1]*d2[1] + d2[2]*d2[2]);
    const float i1 = 1.f / ((n1 == 0.f) ? 1.f : n1);
    const float i2 = 1.f / ((n2 == 0.f) ? 1.f : n2);
    float p1v[3], p2v[3], p12[3];
    #pragma unroll
    for (int k = 0; k < 3; ++k) { p1v[k] = d1[k] * i1; p2v[k] = d2[k] * i2; }
    p12[0] = (p1v[1]*p2v[2] - p1v[2]*p2v[1]) * INV_SQRT6;
    p12[1] = (p1v[2]*p2v[0] - p1v[0]*p2v[2]) * INV_SQRT6;
    p12[2] = (p1v[0]*p2v[1] - p1v[1]*p2v[0]) * INV_SQRT6;

    const size_t base = ((size_t)(b0 + lane) * H_ + h) * 3;
    #pragma unroll
    for (int k = 0; k < 3; ++k) {
      P_out[base + k]  = P[k];
      p1_out[base + k] = p1v[k];
      p2_out[base + k] = p2v[k];
      vecS[wv][lane][k]     = p1v[k];
      vecS[wv][lane][3 + k] = p2v[k];
      vecS[wv][lane][6 + k] = p12[k];
    }
  }
  __syncthreads();

  // rot_axes[b,h,l,k] = p1[k]*Wdec[0,h,l] + p2[k]*Wdec[1,h,l] + p12[k]*Wdec[2,h,l]
  {
    const float* pv = vecS[wv][m];
    const int l0 = (lane < 16) ? 0 : 10;
    #pragma unroll
    for (int li = 0; li < 10; ++li) {
      const int l = l0 + li;
      const float w0 = Wdec[((size_t)0 * H_ + h) * L_ + l];
      const float w1 = Wdec[((size_t)1 * H_ + h) * L_ + l];
      const float w2 = Wdec[((size_t)2 * H_ + h) * L_ + l];
      const size_t ro = (((size_t)(b0 + m) * H_ + h) * L_ + l) * 3;
      rot_out[ro + 0] = pv[0]*w0 + pv[3]*w1 + pv[6]*w2;
      rot_out[ro + 1] = pv[1]*w0 + pv[4]*w1 + pv[7]*w2;
      rot_out[ro + 2] = pv[2]*w0 + pv[5]*w1 + pv[8]*w2;
    }
  }
}

// ---------------------------------------------------------------------------
// Kernel 4: reconstruction as a GATHER (<= 2 windows cover each n), avoiding
// the reference's scatter-add (and float-atomic nondeterminism).
// v3 = q*(0,v)*conj(q) = (w^2 - u.u) v + 2(u.v) u + 2 w (u x v)   (exact algebra)
// ---------------------------------------------------------------------------
__global__ __launch_bounds__(256) void recon_kernel(const float* __restrict__ angles,
                                                    const float* __restrict__ scaling,
                                                    const float* __restrict__ P_out,
                                                    const float* __restrict__ p1_out,
                                                    const float* __restrict__ rot_out,
                                                    const float* __restrict__ nw_out,
                                                    float* __restrict__ recon_out) {
  const long long i = (long long)blockIdx.x * blockDim.x + threadIdx.x;
  if (i >= (long long)B_ * N_) return;
  const int n = (int)(i % N_);
  const int b = (int)(i / N_);

  float a0 = 0.f, a1 = 0.f, a2 = 0.f;
  const int h0 = n / S_;
  #pragma unroll
  for (int d = 0; d < 2; ++d) {
    const int h = h0 - d;
    if (h < 0 || h >= H_) continue;
    const int l = n - S_ * h;            // in [0,10) for d=0, [10,20) for d=1
    const int hl = h * L_ + l;
    const float ang = angles[hl];
    const float sc  = scaling[hl];
    const float cw = cosf(ang);
    const float sw = sinf(ang);
    const size_t rb = (((size_t)b * H_ + h) * L_ + l) * 3;
    const float ux = sw * rot_out[rb + 0];
    const float uy = sw * rot_out[rb + 1];
    const float uz = sw * rot_out[rb + 2];
    const size_t pb = ((size_t)b * H_ + h) * 3;
    const float vx = p1_out[pb + 0], vy = p1_out[pb + 1], vz = p1_out[pb + 2];
    const float w2u  = cw * cw - (ux*ux + uy*uy + uz*uz);
    const float udv2 = 2.f * (ux*vx + uy*vy + uz*vz);
    const float cx = uy*vz - uz*vy;
    const float cy = uz*vx - ux*vz;
    const float cz = ux*vy - uy*vx;
    const float v3x = w2u*vx + udv2*ux + 2.f*cw*cx;
    const float v3y = w2u*vy + udv2*uy + 2.f*cw*cy;
    const float v3z = w2u*vz + udv2*uz + 2.f*cw*cz;
    const float w = nw_out[(size_t)n * H_ + h];
    a0 += (sc * v3x + P_out[pb + 0]) * w;
    a1 += (sc * v3y + P_out[pb + 1]) * w;
    a2 += (sc * v3z + P_out[pb + 2]) * w;
  }
  const size_t ob = ((size_t)b * N_ + n) * 3;
  recon_out[ob + 0] = a0;
  recon_out[ob + 1] = a1;
  recon_out[ob + 2] = a2;
}

// ---------------------------------------------------------------------------
extern "C" void kernel_launch(void* const* d_in, const int* in_sizes, int n_in,
                              void* d_out, int out_size, void* d_ws, size_t ws_size,
                              hipStream_t stream) {
  const float* x       = (const float*)d_in[0];  // (B,N,3)
  const float* Wenc    = (const float*)d_in[1];  // (L,H,3)
  const float* Wdec    = (const float*)d_in[2];  // (3,H,L)
  const float* angles  = (const float*)d_in[3];  // (H*L,1)
  const float* scaling = (const float*)d_in[4];  // (H*L,1)
  float* out = (float*)d_out;

  // Output tuple offsets (flat, in return order)
  const long long oRecon = 0;
  const long long oP     = oRecon + (long long)B_ * N_ * 3;
  const long long oP1    = oP     + (long long)B_ * H_ * 3;
  const long long oP2    = oP1    + (long long)B_ * H_ * 3;
  const long long oSt    = oP2    + (long long)B_ * H_ * 3;
  const long long oRot   = oSt    + (long long)B_ * L_ * H_ * 3;
  const long long oNw    = oRot   + (long long)B_ * H_ * L_ * 3;
  const long long oC     = oNw    + (long long)N_ * H_;

  // 1) nw + contribution (8 rows per 256-thread block)
  nw_kernel<<<(N_ + 7) / 8, 256, 0, stream>>>(Wenc, out + oNw, out + oC);

  // 2) st gather
  const long long totSt = (long long)B_ * L_ * H_ * 3;
  st_kernel<<<(unsigned)((totSt + 255) / 256), 256, 0, stream>>>(x, out + oSt);

  // 3) WMMA encode: 32 b-tiles * H waves, 8 waves/block -> exactly 4004 blocks
  encode_kernel<<<(32 * H_) / 8, 256, 0, stream>>>(x, Wenc, Wdec,
                                                   out + oP, out + oP1,
                                                   out + oP2, out + oRot);

  // 4) reconstruction gather (reads outputs of kernels 1 & 3; same stream)
  const long long totR = (long long)B_ * N_;
  recon_kernel<<<(unsigned)((totR + 255) / 256), 256, 0, stream>>>(
      angles, scaling, out + oP, out + oP1, out + oRot, out + oNw, out + oRecon);
}